// PredictedSequenceFreqHistLayer_66460323938629
// MI455X (gfx1250) — compile-verified
//
#include <hip/hip_runtime.h>
#include <math.h>

#define VOCAB   32000
#define BATCH   4
#define SEQLEN  2048
#define THREADS 256
#define WAVES   (THREADS / 32)

// Async global->LDS staging path (gfx1250): guarded so the file compiles on
// host pass and on toolchains lacking the builtins.
#if defined(__gfx1250__) && defined(__HIP_DEVICE_COMPILE__) && \
    __has_builtin(__builtin_amdgcn_global_load_async_to_lds_b128) && \
    __has_builtin(__builtin_amdgcn_s_wait_asynccnt)
#define USE_ASYNC_LDS 1
#else
#define USE_ASYNC_LDS 0
#endif

#if USE_ASYNC_LDS
typedef int v4i __attribute__((__vector_size__(4 * sizeof(int))));
typedef __attribute__((address_space(1))) v4i as1_v4i;   // prints as "v4i __device__ *"
typedef __attribute__((address_space(3))) v4i as3_v4i;   // prints as "v4i __shared__ *"
#endif

__global__ void hist_zero_kernel(float* __restrict__ out, int n) {
    int i = blockIdx.x * blockDim.x + threadIdx.x;
    if (i < n) out[i] = 0.0f;
}

// One block per (batch, seq) row: argmax over VOCAB f32, then one atomic
// increment into the per-batch histogram.
__global__ __launch_bounds__(THREADS) void argmax_hist_kernel(
    const float* __restrict__ logits, float* __restrict__ out) {
    const int s = blockIdx.x;       // 0..SEQLEN-1
    const int b = blockIdx.y;       // 0..BATCH-1
    const int tid = threadIdx.x;
    const float* row = logits + ((size_t)b * SEQLEN + s) * VOCAB;

    float bestV = -INFINITY;
    int   bestI = 0x7fffffff;

#if USE_ASYNC_LDS
    // Stream the row through LDS in 1024-float (4 KB) tiles, double-buffered.
    // Each lane stages and consumes its own 16 B, so only the per-wave
    // ASYNCcnt wait is required (loads complete in order) — no barriers.
    __shared__ float tilebuf[2 * 1024];
    constexpr int NTILE = 31;                 // 31*1024 = 31744 floats
    {
        as1_v4i* g = (as1_v4i*)(row + tid * 4);
        as3_v4i* l = (as3_v4i*)(&tilebuf[tid * 4]);
        __builtin_amdgcn_global_load_async_to_lds_b128(g, l, 0, 0);
    }
    for (int t = 0; t < NTILE; ++t) {
        if (t + 1 < NTILE) {
            as1_v4i* g = (as1_v4i*)(row + (t + 1) * 1024 + tid * 4);
            as3_v4i* l = (as3_v4i*)(&tilebuf[((t + 1) & 1) * 1024 + tid * 4]);
            __builtin_amdgcn_global_load_async_to_lds_b128(g, l, 0, 0);
            __builtin_amdgcn_s_wait_asynccnt(1);   // tile t landed; t+1 in flight
        } else {
            __builtin_amdgcn_s_wait_asynccnt(0);   // last tile landed
        }
        const float4 v = *(const float4*)(&tilebuf[(t & 1) * 1024 + tid * 4]);
        const int base = t * 1024 + tid * 4;       // indices increase => '>' keeps first max
        if (v.x > bestV) { bestV = v.x; bestI = base; }
        if (v.y > bestV) { bestV = v.y; bestI = base + 1; }
        if (v.z > bestV) { bestV = v.z; bestI = base + 2; }
        if (v.w > bestV) { bestV = v.w; bestI = base + 3; }
    }
    {   // tail: 32000 - 31744 = 256 floats, one per thread
        const int idx = NTILE * 1024 + tid;
        const float v = row[idx];
        if (v > bestV || (v == bestV && idx < bestI)) { bestV = v; bestI = idx; }
    }
#else
    // Fallback: direct 128-bit coalesced streaming loads.
    const float4* row4 = (const float4*)row;
    for (int i = tid; i < VOCAB / 4; i += THREADS) {
        const float4 v = row4[i];
        const int base = i * 4;                    // i strictly increases per thread
        if (v.x > bestV) { bestV = v.x; bestI = base; }
        if (v.y > bestV) { bestV = v.y; bestI = base + 1; }
        if (v.z > bestV) { bestV = v.z; bestI = base + 2; }
        if (v.w > bestV) { bestV = v.w; bestI = base + 3; }
    }
#endif

    // wave32 (val, idx) reduction; tie -> lower index (first occurrence)
    for (int off = 16; off > 0; off >>= 1) {
        const float ov = __shfl_xor(bestV, off, 32);
        const int   oi = __shfl_xor(bestI, off, 32);
        if (ov > bestV || (ov == bestV && oi < bestI)) { bestV = ov; bestI = oi; }
    }

    __shared__ float sv[WAVES];
    __shared__ int   si[WAVES];
    const int wave = tid >> 5;
    if ((tid & 31) == 0) { sv[wave] = bestV; si[wave] = bestI; }
    __syncthreads();

    if (tid == 0) {
        float v = sv[0];
        int   ix = si[0];
        for (int w = 1; w < WAVES; ++w) {
            if (sv[w] > v || (sv[w] == v && si[w] < ix)) { v = sv[w]; ix = si[w]; }
        }
        atomicAdd(&out[(size_t)b * VOCAB + ix], 1.0f);
    }
}

// In-place: zero special tokens {0,1,2}, clamp to MAX_COPIES=4, normalize.
__global__ void hist_finalize_kernel(float* __restrict__ out, int n) {
    int i = blockIdx.x * blockDim.x + threadIdx.x;
    if (i >= n) return;
    const int v = i % VOCAB;
    const float c = out[i];
    out[i] = (v < 3) ? 0.0f : fminf(c, 4.0f) * 0.25f;
}

extern "C" void kernel_launch(void* const* d_in, const int* in_sizes, int n_in,
                              void* d_out, int out_size, void* d_ws, size_t ws_size,
                              hipStream_t stream) {
    (void)in_sizes; (void)n_in; (void)d_ws; (void)ws_size; (void)out_size;
    const float* logits = (const float*)d_in[0];
    float* out = (float*)d_out;

    const int n = BATCH * VOCAB;
    hist_zero_kernel<<<(n + 255) / 256, 256, 0, stream>>>(out, n);

    dim3 grid(SEQLEN, BATCH);
    argmax_hist_kernel<<<grid, THREADS, 0, stream>>>(logits, out);

    hist_finalize_kernel<<<(n + 255) / 256, 256, 0, stream>>>(out, n);
}